// RTDETRPostProcessor_8117488190270
// MI455X (gfx1250) — compile-verified
//
#include <hip/hip_runtime.h>
#include <hip/hip_bf16.h>
#include <stdint.h>

#define NCLS        80       // classes per query (reference constant)
#define QTILE       64       // queries per block in kernel 1
#define K1_THREADS  256      // 8 wave32s
#define THRESH      0.05f
#define MAXDET      300
#define SORT_N      1024
#define K2_THREADS  512      // SORT_N/2 compare pairs, 16 wave32s

typedef int v4i __attribute__((ext_vector_type(4)));
typedef unsigned int u32x4 __attribute__((ext_vector_type(4)));
typedef int i32x8 __attribute__((ext_vector_type(8)));
typedef int i32x4 __attribute__((ext_vector_type(4)));
typedef __attribute__((address_space(1))) v4i* g_v4i_p;   // global (AS1) int4*
typedef __attribute__((address_space(3))) v4i* l_v4i_p;   // LDS (AS3) int4*
typedef __attribute__((address_space(3))) void as3_void;

// ---- CDNA5 async global->LDS (ASYNCcnt path) ----
__device__ __forceinline__ void async_ld_b128(const void* g, void* l) {
#if __has_builtin(__builtin_amdgcn_global_load_async_to_lds_b128)
    __builtin_amdgcn_global_load_async_to_lds_b128((g_v4i_p)g, (l_v4i_p)l, 0, 0);
#else
    unsigned loff = (unsigned)(uintptr_t)(as3_void*)l;     // 32-bit LDS byte offset
    unsigned long long ga = (unsigned long long)(uintptr_t)g;
    asm volatile("global_load_async_to_lds_b128 %0, %1, off" :: "v"(loff), "v"(ga) : "memory");
#endif
}

__device__ __forceinline__ void wait_async0() {
#if __has_builtin(__builtin_amdgcn_s_wait_asynccnt)
    __builtin_amdgcn_s_wait_asynccnt(0);
#else
    asm volatile("s_wait_asynccnt 0" ::: "memory");
#endif
}

__device__ __forceinline__ void wait_tensor0() {
#if __has_builtin(__builtin_amdgcn_s_wait_tensorcnt)
    __builtin_amdgcn_s_wait_tensorcnt(0);
#else
    asm volatile("s_wait_tensorcnt 0" ::: "memory");
#endif
}

// ============================================================================
// Kernel 1: per-query max/argmax over 80 logits; sigmoid(max) == max(sigmoid).
// TDM DMAs a flat 64x80 f32 tile (20KB) into LDS in ONE instruction; waves
// sleep on TENSORcnt, then 4 lanes/query reduce.
// ============================================================================
__global__ __launch_bounds__(K1_THREADS)
void score_argmax_kernel(const float* __restrict__ logits,
                         float* __restrict__ ws_scores,
                         int*   __restrict__ ws_labels,
                         int total_q)
{
    __shared__ float smem[QTILE * NCLS];       // 20480 B, flat 1-D tile
    const int tid = threadIdx.x;
    const long long qbase = (long long)blockIdx.x * QTILE;

#if __has_builtin(__builtin_amdgcn_tensor_load_to_lds)
    // ---- Tensor Data Mover: one descriptor moves the whole tile ----
    if (tid < 32) {                            // wave 0 only issues the DMA
        long long rem = (total_q - qbase) * (long long)NCLS;
        unsigned elems = (unsigned)(rem < (long long)(QTILE * NCLS) ? rem
                                                                    : (QTILE * NCLS)); // 5120
        unsigned lds_base = (unsigned)(uintptr_t)(as3_void*)&smem[0];
        unsigned long long ga =
            (unsigned long long)(uintptr_t)(logits + qbase * NCLS);

        // D# group 0: count=1 | lds_addr | global_addr[56:0] | type=2 ("image")
        u32x4 g0 = { 1u,
                     lds_base,
                     (unsigned)ga,
                     (unsigned)((ga >> 32) & 0x01FFFFFFu) | 0x80000000u };
        // D# group 1: data_size=4B(code2); tensor_dim0=tile_dim0=elems (1-D copy);
        //             tensor_dim1=1, tile_dim1=1; stride0=elems
        i32x8 g1 = { (int)(2u << 16),                               // dw0: data_size
                     (int)((elems & 0xFFFFu) << 16),                // dw1: tensor_dim0[15:0]
                     (int)(((elems >> 16) & 0xFFFFu) | (1u << 16)), // dw2: tensor_dim0[31:16], tensor_dim1=1
                     (int)((elems & 0xFFFFu) << 16),                // dw3: tensor_dim1[31:16]=0, tile_dim0
                     1,                                             // dw4: tile_dim1=1, tile_dim2=0
                     (int)elems,                                    // dw5: tensor_dim0_stride[31:0]
                     0,                                             // dw6: stride0 hi, stride1 lo
                     0 };                                           // dw7: stride1 hi
        i32x4 g2 = { 0, 0, 0, 0 };             // <=2-D: no dims 2..4
        i32x4 g3 = { 0, 0, 0, 0 };
        i32x8 g4 = { 0, 0, 0, 0, 0, 0, 0, 0 }; // 6-arg toolchain form: extra group, zeroed
        __builtin_amdgcn_tensor_load_to_lds(g0, g1, g2, g3, g4, 0);
        wait_tensor0();                        // TENSORcnt -> 0 before the barrier
    }
    __syncthreads();
#else
    // ---- fallback: per-lane async-DMA stage, fully coalesced b128s ----
    const long long gchunk_base = qbase * (NCLS / 4);   // global float4 index
    #pragma unroll
    for (int k = 0; k < (QTILE * NCLS / 4) / K1_THREADS; ++k) {   // 5 iters
        int f  = tid + k * K1_THREADS;       // 0..1279
        int q  = f / (NCLS / 4);
        int e4 = f % (NCLS / 4);
        if (qbase + q < total_q) {
            const float* gsrc = logits + (gchunk_base + f) * 4;
            float*       ldst = &smem[q * NCLS + e4 * 4];
            async_ld_b128(gsrc, ldst);
        }
    }
    wait_async0();
    __syncthreads();
#endif

    // ---- 4 lanes per query, 20 classes each; first-occurrence argmax ----
    const int q    = tid >> 2;
    const int part = tid & 3;
    const float* row = &smem[q * NCLS + part * 20];
    float best  = row[0];
    int   bestc = part * 20;
    #pragma unroll
    for (int j = 1; j < 20; ++j) {
        float v = row[j];
        if (v > best) { best = v; bestc = part * 20 + j; }
    }
    // wave32 butterfly combine; ties -> lowest class index (matches jnp.argmax)
    #pragma unroll
    for (int m = 1; m <= 2; m <<= 1) {
        float ov = __shfl_xor(best, m, 32);
        int   oc = __shfl_xor(bestc, m, 32);
        if (ov > best || (ov == best && oc < bestc)) { best = ov; bestc = oc; }
    }

    const long long gq = qbase + q;
    if (part == 0 && gq < total_q) {
        float s = 1.0f / (1.0f + __expf(-best));        // sigmoid(max logit)
        ws_scores[gq] = (s > THRESH) ? s : -1.0f;       // pre-masked (reference semantics)
        ws_labels[gq] = bestc;
    }
}

// ============================================================================
// Kernel 2: per-row top-300 of 1000 via LDS bitonic sort (descending, ties ->
// lower index first, matching lax.top_k), then gather + box transform.
// ============================================================================
__global__ __launch_bounds__(K2_THREADS)
void topk_gather_kernel(const float*  __restrict__ ws_scores,
                        const int*    __restrict__ ws_labels,
                        const float4* __restrict__ boxes,
                        const float*  __restrict__ tsizes,
                        float* __restrict__ out_scores,
                        float* __restrict__ out_labels,
                        float* __restrict__ out_boxes,
                        int Q)
{
    __shared__ float vals[SORT_N];
    __shared__ int   idxs[SORT_N];
    const int b = blockIdx.x;
    const int t = threadIdx.x;

    for (int e = t; e < SORT_N; e += K2_THREADS) {
        vals[e] = (e < Q) ? ws_scores[(long long)b * Q + e] : -3.0e38f;
        idxs[e] = e;
    }

    for (int k = 2; k <= SORT_N; k <<= 1) {
        for (int j = k >> 1; j > 0; j >>= 1) {
            __syncthreads();
            int i = ((t & ~(j - 1)) << 1) | (t & (j - 1));
            int p = i | j;
            float v1 = vals[i], v2 = vals[p];
            int   a1 = idxs[i], a2 = idxs[p];
            bool firstBefore = (v1 > v2) || (v1 == v2 && a1 < a2); // desired order
            bool up = ((i & k) == 0);
            if (up ? !firstBefore : firstBefore) {
                vals[i] = v2; vals[p] = v1;
                idxs[i] = a2; idxs[p] = a1;
            }
        }
    }
    __syncthreads();

    if (t < MAXDET) {
        float v = vals[t];
        int   q = idxs[t];
        bool valid = v > THRESH;

        out_scores[b * MAXDET + t] = valid ? v : 0.0f;
        out_labels[b * MAXDET + t] =
            valid ? (float)ws_labels[(long long)b * Q + q] : -1.0f;

        float4 bx = boxes[(long long)b * Q + q];   // (cx, cy, w, h)
        float ih = tsizes[2 * b + 0];
        float iw = tsizes[2 * b + 1];
        float x0 = (bx.x - 0.5f * bx.z) * iw;
        float y0 = (bx.y - 0.5f * bx.w) * ih;
        float x1 = (bx.x + 0.5f * bx.z) * iw;
        float y1 = (bx.y + 0.5f * bx.w) * ih;

        long long o = ((long long)b * MAXDET + t) * 4;
        out_boxes[o + 0] = valid ? x0 : 0.0f;
        out_boxes[o + 1] = valid ? y0 : 0.0f;
        out_boxes[o + 2] = valid ? x1 : 0.0f;
        out_boxes[o + 3] = valid ? y1 : 0.0f;
    }
}

// ============================================================================
extern "C" void kernel_launch(void* const* d_in, const int* in_sizes, int n_in,
                              void* d_out, int out_size, void* d_ws, size_t ws_size,
                              hipStream_t stream)
{
    (void)n_in; (void)out_size; (void)ws_size;

    const float* logits = (const float*)d_in[0];   // (B, Q, 80)
    const float* pboxes = (const float*)d_in[1];   // (B, Q, 4)
    const float* tsz    = (const float*)d_in[2];   // (B, 2)  [h, w]

    const int B = in_sizes[2] / 2;                 // 256
    const int Q = in_sizes[1] / (4 * B);           // 1000
    const int total_q = B * Q;

    // workspace: masked scores + labels per query
    float* ws_scores = (float*)d_ws;
    int*   ws_labels = (int*)(ws_scores + total_q);

    // outputs concatenated flat in return order (all as float)
    float* out_scores = (float*)d_out;                      // B*300
    float* out_labels = out_scores + (size_t)B * MAXDET;    // B*300
    float* out_boxes  = out_labels + (size_t)B * MAXDET;    // B*300*4

    const int blocks1 = (total_q + QTILE - 1) / QTILE;      // 4000
    score_argmax_kernel<<<blocks1, K1_THREADS, 0, stream>>>(
        logits, ws_scores, ws_labels, total_q);

    topk_gather_kernel<<<B, K2_THREADS, 0, stream>>>(
        ws_scores, ws_labels, (const float4*)pboxes, tsz,
        out_scores, out_labels, out_boxes, Q);
}